// CrossNetwork_91242285237049
// MI455X (gfx1250) — compile-verified
//
#include <hip/hip_runtime.h>
#include <hip/hip_bf16.h>
#include <math.h>

// ---------------------------------------------------------------------------
// CrossNetwork on MI455X (gfx1250), fp32 WMMA path (V_WMMA_F32_16X16X4_F32).
// B=8192, T=2, N=40, D=64, H=1, NMASK=3
// ---------------------------------------------------------------------------

#define CN_B 8192
#define CN_T 2
#define CN_N 40
#define CN_D 64
#define CN_F 10240              // T*N*2D
#define CN_EPS 1e-5f

#define PAD_N 48                // N padded to 3x16 WMMA tiles
#define SA 52                   // At row stride (52 mod 64 -> conflict-free A-frag reads)
#define SH 66                   // h/nei/hout row stride (66 mod 64 = 2 -> conflict-free)

#define WPACK_FLOATS (CN_T * 2 * CN_D * CN_D)   // 16384: repacked gnn_w
#define GATEWT_FLOATS (CN_N * CN_N * CN_D)      // 102400: transposed gate_w

typedef float v2f __attribute__((ext_vector_type(2)));
typedef float v8f __attribute__((ext_vector_type(8)));

// ---------------------------------------------------------------------------
// Kernel 1: adjacency pipeline -> A^T padded [T][48][52] in workspace.
// ---------------------------------------------------------------------------
__global__ void cn_prep_adj(const float* __restrict__ masker,
                            const float* __restrict__ ln_gamma,
                            const float* __restrict__ ln_beta,
                            float* __restrict__ At) {
    const int t = blockIdx.x;
    const int n = threadIdx.x;
    if (n >= PAD_N) return;
    float* Arow = At + (t * PAD_N + n) * SA;
    if (n >= CN_N) {                       // zero pad rows
        for (int m = 0; m < SA; ++m) Arow[m] = 0.f;
        return;
    }
    const float* mk = masker + t * 3 * CN_N * CN_N;
    float adj[CN_N];
    float mu = 0.f;
#pragma unroll
    for (int m = 0; m < CN_N; ++m) {
        float p = mk[(0 * CN_N + m) * CN_N + n]
                * mk[(1 * CN_N + m) * CN_N + n]
                * mk[(2 * CN_N + m) * CN_N + n];
        p = fmaxf(p, 0.f);                 // relu(prod)
        adj[m] = p;
        mu += p;
    }
    mu *= (1.f / CN_N);
    float var = 0.f;
#pragma unroll
    for (int m = 0; m < CN_N; ++m) { float d = adj[m] - mu; var += d * d; }
    var *= (1.f / CN_N);
    const float rstd = rsqrtf(var + CN_EPS);

    float xm[CN_N];
    float mx = -3.4e38f;
#pragma unroll
    for (int m = 0; m < CN_N; ++m) {
        float ln = (adj[m] - mu) * rstd * ln_gamma[m] + ln_beta[m];
        float v  = ln + ((adj[m] != 0.f) ? 0.f : -1e9f) + ((m == n) ? 1.f : 0.f);
        xm[m] = v;
        mx = fmaxf(mx, v);
    }
    float s = 0.f;
#pragma unroll
    for (int m = 0; m < CN_N; ++m) { xm[m] = expf(xm[m] - mx); s += xm[m]; }
    const float inv = 1.f / s;
#pragma unroll
    for (int m = 0; m < CN_N; ++m)
        Arow[m] = (adj[m] != 0.f) ? xm[m] * inv : 0.f;   // softmax * mask
    for (int m = CN_N; m < SA; ++m) Arow[m] = 0.f;       // zero K-pad
}

// ---------------------------------------------------------------------------
// Kernel 2: fold BatchNorm (H=1) into scale/shift.
// ---------------------------------------------------------------------------
__global__ void cn_prep_bn(const float* __restrict__ g, const float* __restrict__ b,
                           const float* __restrict__ mean, const float* __restrict__ var,
                           float* __restrict__ scale, float* __restrict__ shift) {
    int i = blockIdx.x * blockDim.x + threadIdx.x;
    if (i >= CN_F) return;
    float sc = g[i] * rsqrtf(var[i] + CN_EPS);
    scale[i] = sc;
    shift[i] = b[i] - mean[i] * sc;
}

// ---------------------------------------------------------------------------
// Kernel 3: repack gnn_w into WMMA B-fragment order (float2 per lane-load)
//   wpack[(((t*32 + k0/4)*2 + koff/2)*64 + col)*2 + r] = W[t][4*(k0/4)+koff+r][col]
// and transpose gate_w -> gate_wT[n][e] for coalesced gate dot-products.
// ---------------------------------------------------------------------------
__global__ void cn_prep_w(const float* __restrict__ gnn_w,
                          const float* __restrict__ gate_w,
                          float* __restrict__ wpack,
                          float* __restrict__ gate_wT) {
    int i = blockIdx.x * blockDim.x + threadIdx.x;
    if (i < WPACK_FLOATS) {
        const int r  = i & 1;
        const int c  = (i >> 1) & 63;
        const int p  = (i >> 7) & 1;
        const int kq = (i >> 8) & 31;
        const int t  = i >> 13;
        const int k  = 4 * kq + 2 * p + r;
        wpack[i] = gnn_w[(size_t)t * 2 * CN_D * CN_D + k * CN_D + c];
    } else {
        int j = i - WPACK_FLOATS;
        if (j < GATEWT_FLOATS) {
            const int n = j / (CN_N * CN_D);
            const int e = j - n * (CN_N * CN_D);
            gate_wT[j] = gate_w[e * CN_N + n];
        }
    }
}

// ---------------------------------------------------------------------------
// Main kernel: one workgroup (8 wave32) per (b, t).
// ---------------------------------------------------------------------------
__global__ __launch_bounds__(256)
void cn_main(const float* __restrict__ x,         // (B,N,D)
             const float* __restrict__ wpack,     // ws: frag-packed gnn_w
             const float* __restrict__ gnn_b,     // (T,1,N,D)
             const float* __restrict__ gate_wT,   // ws: (N, N*D) transposed
             const float* __restrict__ gate_b,    // (N,)
             const float* __restrict__ At,        // ws: (T,48,52)
             const float* __restrict__ bn_scale,  // ws: (F,)
             const float* __restrict__ bn_shift,  // ws: (F,)
             float* __restrict__ out) {           // (B,T,D)
    __shared__ float h_s  [PAD_N * SH];
    __shared__ float nei_s[PAD_N * SH];
    __shared__ float hout_s[PAD_N * SH];
    __shared__ float At_s [PAD_N * SA];
    __shared__ float logits_s[CN_N];

    const int b    = blockIdx.x;
    const int t    = blockIdx.y;
    const int tid  = threadIdx.x;
    const int wave = tid >> 5;
    const int lane = tid & 31;
    const int l16  = lane & 15;
    const int koff = (lane < 16) ? 0 : 2;  // K split for fp32 A/B fragments
    const int half = (lane < 16) ? 0 : 8;  // M split for C/D fragments

    // ---- stage h (pad rows zero) and A^T into LDS -------------------------
    for (int i = tid; i < 8 * SH; i += 256) h_s[CN_N * SH + i] = 0.f;  // rows 40..47
    const float* xb = x + (size_t)b * CN_N * CN_D;
    for (int i = tid; i < CN_N * CN_D; i += 256)
        h_s[(i >> 6) * SH + (i & 63)] = xb[i];
    for (int i = tid; i < PAD_N * SA; i += 256)
        At_s[i] = At[t * PAD_N * SA + i];
    __syncthreads();

    // ---- nei = A^T @ h : M=48(n) K=48(m) N=64(d), fp32 WMMA ---------------
    for (int tile = wave; tile < 12; tile += 8) {
        const int Mb = (tile >> 2) * 16;   // n tile
        const int Nb = (tile & 3) * 16;    // d tile
        v8f acc = {};
#pragma unroll
        for (int k0 = 0; k0 < PAD_N; k0 += 4) {
            v2f a, bb;
            const float* ap = At_s + (Mb + l16) * SA + k0 + koff;
            a.x = ap[0];
            a.y = ap[1];
            bb.x = h_s[(k0 + koff    ) * SH + Nb + l16];
            bb.y = h_s[(k0 + koff + 1) * SH + Nb + l16];
            acc = __builtin_amdgcn_wmma_f32_16x16x4_f32(
                false, a, false, bb, (short)0, acc, false, false);
        }
#pragma unroll
        for (int r = 0; r < 8; ++r)
            nei_s[(Mb + r + half) * SH + Nb + l16] = acc[r];
    }
    __syncthreads();

    // ---- BatchNorm affine in place on hc = [h | nei] (rows < 40) ----------
    const float* sc_t = bn_scale + t * CN_N * 2 * CN_D;
    const float* sh_t = bn_shift + t * CN_N * 2 * CN_D;
    for (int i = tid; i < CN_N * 2 * CN_D; i += 256) {
        const int n = i >> 7;
        const int f = i & 127;
        float* p = (f < CN_D) ? (h_s + n * SH + f) : (nei_s + n * SH + (f - CN_D));
        *p = *p * sc_t[i] + sh_t[i];
    }
    __syncthreads();

    // ---- h' = hc @ W_t + bias : M=48 K=128 N=64 ---------------------------
    // B-fragments come from the frag-packed W: one float2 (b64) per K-step.
    const float2* Wp = (const float2*)wpack;   // [(t*32+kq)*2+p][64] float2
    const float*  Bt = gnn_b + (size_t)t * CN_N * CN_D;   // (40,64)
    for (int tile = wave; tile < 12; tile += 8) {
        const int Mb = (tile >> 2) * 16;
        const int Nb = (tile & 3) * 16;
        v8f acc = {};
#pragma unroll
        for (int k0 = 0; k0 < 2 * CN_D; k0 += 4) {
            const int f0 = k0 + koff;
            const float* ap = (f0 < CN_D)
                ? (h_s   + (Mb + l16) * SH + f0)
                : (nei_s + (Mb + l16) * SH + (f0 - CN_D));
            v2f a, bb;
            a.x = ap[0];
            a.y = ap[1];
            const float2 wv =
                Wp[((t * 32 + (k0 >> 2)) * 2 + (koff >> 1)) * CN_D + Nb + l16];
            bb.x = wv.x;
            bb.y = wv.y;
            acc = __builtin_amdgcn_wmma_f32_16x16x4_f32(
                false, a, false, bb, (short)0, acc, false, false);
        }
#pragma unroll
        for (int r = 0; r < 8; ++r) {
            const int row = Mb + r + half;
            const int col = Nb + l16;
            float v = acc[r];
            if (row < CN_N) v += Bt[row * CN_D + col];
            hout_s[row * SH + col] = v;
        }
    }
    __syncthreads();

    // ---- gate logits: logits[n] = h_flat . gate_wT[n,:] + gate_b[n] -------
    // wave w handles n = w, w+8, ..., w+32 (5 logits); coalesced 32-lane dot.
#pragma unroll
    for (int j = 0; j < 5; ++j) {
        const int n = wave + 8 * j;
        const float* gwn = gate_wT + (size_t)n * CN_N * CN_D;
        float partial = 0.f;
        for (int e = lane; e < CN_N * CN_D; e += 32)
            partial += hout_s[(e >> 6) * SH + (e & 63)] * gwn[e];
#pragma unroll
        for (int off = 16; off > 0; off >>= 1)
            partial += __shfl_xor(partial, off, 32);
        if (lane == 0) logits_s[n] = partial + gate_b[n];
    }
    __syncthreads();

    // ---- softmax over fields + weighted pool -> out[b][t][d] --------------
    if (tid < CN_D) {
        const int d = tid;
        float mx = -3.4e38f;
#pragma unroll
        for (int n = 0; n < CN_N; ++n) mx = fmaxf(mx, logits_s[n]);
        float s = 0.f;
        float acc = 0.f;
#pragma unroll
        for (int n = 0; n < CN_N; ++n) {
            float e = expf(logits_s[n] - mx);
            s += e;
            acc += e * hout_s[n * SH + d];
        }
        out[((size_t)b * CN_T + t) * CN_D + d] = acc / s;
    }
}

// ---------------------------------------------------------------------------
// Host launcher
// ---------------------------------------------------------------------------
extern "C" void kernel_launch(void* const* d_in, const int* in_sizes, int n_in,
                              void* d_out, int out_size, void* d_ws, size_t ws_size,
                              hipStream_t stream) {
    const float* x        = (const float*)d_in[0];
    const float* masker   = (const float*)d_in[1];
    const float* ln_gamma = (const float*)d_in[2];
    const float* ln_beta  = (const float*)d_in[3];
    const float* gnn_w    = (const float*)d_in[4];
    const float* gnn_b    = (const float*)d_in[5];
    const float* bn_gamma = (const float*)d_in[6];
    const float* bn_beta  = (const float*)d_in[7];
    const float* bn_mean  = (const float*)d_in[8];
    const float* bn_var   = (const float*)d_in[9];
    const float* gate_w   = (const float*)d_in[10];
    const float* gate_b   = (const float*)d_in[11];
    float* out = (float*)d_out;

    // workspace layout (floats):
    //   At[2*48*52] | bn_scale[F] | bn_shift[F] | wpack[16384] | gate_wT[102400]
    float* ws       = (float*)d_ws;
    float* At       = ws;
    float* bn_scale = At + CN_T * PAD_N * SA;
    float* bn_shift = bn_scale + CN_F;
    float* wpack    = bn_shift + CN_F;
    float* gate_wT  = wpack + WPACK_FLOATS;

    cn_prep_adj<<<dim3(CN_T), 64, 0, stream>>>(masker, ln_gamma, ln_beta, At);
    cn_prep_bn<<<dim3((CN_F + 255) / 256), 256, 0, stream>>>(
        bn_gamma, bn_beta, bn_mean, bn_var, bn_scale, bn_shift);
    cn_prep_w<<<dim3((WPACK_FLOATS + GATEWT_FLOATS + 255) / 256), 256, 0, stream>>>(
        gnn_w, gate_w, wpack, gate_wT);
    cn_main<<<dim3(CN_B, CN_T), 256, 0, stream>>>(
        x, wpack, gnn_b, gate_wT, gate_b, At, bn_scale, bn_shift, out);
}